// KnnMemory_46110768890086
// MI455X (gfx1250) — compile-verified
//
#include <hip/hip_runtime.h>
#include <hip/hip_bf16.h>

typedef float v2f __attribute__((ext_vector_type(2)));
typedef float v8f __attribute__((ext_vector_type(8)));

#define DIM   512
#define KCOLS 8192
#define TOPK  100
#define NROWS 8192           // B*N = 16*512
#define WAVES 8              // waves per gemm workgroup (wave32)
#define CTILES 4             // independent 16x16 column tiles per wave
#define XPITCH 516           // padded LDS row stride (516 mod 64 = 4 -> conflict-free)

// ---------------------------------------------------------------------------
// Transpose queue [DIM, K] (row-major) -> qT [K, DIM] so gathered feature rows
// are contiguous 2KB reads (L2-resident, 192MB L2 >> 16MB).
// ---------------------------------------------------------------------------
__global__ __launch_bounds__(256) void knn_transpose(const float* __restrict__ q,
                                                     float* __restrict__ qT) {
    __shared__ float tile[32][33];
    int kx = blockIdx.x * 32 + threadIdx.x;   // K index
    int dy = blockIdx.y * 32 + threadIdx.y;   // DIM index
#pragma unroll
    for (int j = 0; j < 32; j += 8)
        tile[threadIdx.y + j][threadIdx.x] = q[(size_t)(dy + j) * KCOLS + kx];
    __syncthreads();
    int dx = blockIdx.y * 32 + threadIdx.x;   // DIM index (output col)
    int ky = blockIdx.x * 32 + threadIdx.y;   // K index   (output row)
#pragma unroll
    for (int j = 0; j < 32; j += 8)
        qT[(size_t)(ky + j) * DIM + dx] = tile[threadIdx.x][threadIdx.y + j];
}

// ---------------------------------------------------------------------------
// GEMM: sim[rows, K] = X[rows, DIM] * Q[DIM, K] using V_WMMA_F32_16X16X4_F32.
// Block = 256 threads = 8 waves. Each wave owns CTILES=4 independent 16x16
// column tiles (64 cols), so the block covers a 16-row x 512-col output slab.
// The 4 accumulator chains are independent -> WMMA pipeline stays fed, and one
// A operand (LDS) feeds 4 WMMAs.
//
// f32 A-matrix 16x4 layout (ISA 7.12.2): lane&15 = M, VGPR0 holds K=(lane>>4)*2,
// VGPR1 holds K+1.  B-matrix 4x16 mirrors with lane&15 = N.
// C/D 16x16: VGPR r -> M = r + 8*(lane>>4), N = lane&15.
// ---------------------------------------------------------------------------
__global__ __launch_bounds__(256) void knn_gemm(const float* __restrict__ x,
                                                const float* __restrict__ q,
                                                float* __restrict__ sim,
                                                int row_base) {
    __shared__ float xs[16 * XPITCH];         // ~33KB, padded rows

    // Cooperative load of the 16-row X tile into padded LDS (float4 bursts).
    {
        const float4* src = (const float4*)(x + (size_t)(row_base + blockIdx.x * 16) * DIM);
        float4* dst = (float4*)xs;
        const int n4 = 16 * DIM / 4;          // 2048 float4s
        for (int i = threadIdx.x; i < n4; i += 256) {
            int row  = i >> 7;                // /128 float4s per source row
            int col4 = i & 127;
            // 516 floats = 129 float4s of row pitch; row base stays 16B aligned.
            dst[row * (XPITCH / 4) + col4] = src[i];
        }
    }
    __syncthreads();

    const int wave = threadIdx.x >> 5;
    const int lane = threadIdx.x & 31;
    const int col_base = (blockIdx.y * WAVES + wave) * (16 * CTILES);

    const int mn = lane & 15;                 // M for A, N for B
    const int kk = (lane >> 4) * 2;           // K sub-offset per ISA layout

    v8f c0 = {0.f,0.f,0.f,0.f,0.f,0.f,0.f,0.f};
    v8f c1 = c0, c2 = c0, c3 = c0;

    const float* xrow = xs + mn * XPITCH;     // A source (LDS, conflict-free)
    const float* qcol = q + col_base + mn;    // B source (global, L2-resident)

#pragma unroll 2
    for (int k0 = 0; k0 < DIM; k0 += 4) {
        v2f a = *(const v2f*)(xrow + k0 + kk);          // one ds_load_b64
        const float* qp = qcol + (size_t)(k0 + kk) * KCOLS;
        v2f b0, b1, b2, b3;
        b0.x = qp[0];      b0.y = qp[KCOLS];
        b1.x = qp[16];     b1.y = qp[KCOLS + 16];
        b2.x = qp[32];     b2.y = qp[KCOLS + 32];
        b3.x = qp[48];     b3.y = qp[KCOLS + 48];
        // 8 args: (neg_a, A, neg_b, B, c_mod, C, reuse_a, reuse_b)
        c0 = __builtin_amdgcn_wmma_f32_16x16x4_f32(false, a, false, b0, (short)0, c0, false, false);
        c1 = __builtin_amdgcn_wmma_f32_16x16x4_f32(false, a, false, b1, (short)0, c1, false, false);
        c2 = __builtin_amdgcn_wmma_f32_16x16x4_f32(false, a, false, b2, (short)0, c2, false, false);
        c3 = __builtin_amdgcn_wmma_f32_16x16x4_f32(false, a, false, b3, (short)0, c3, false, false);
    }

    // Write back per C/D layout. sim is chunk-local: row index = local row.
    const int colw = col_base + (lane & 15);
    const int mrow = blockIdx.x * 16 + (lane >> 4) * 8;
#pragma unroll
    for (int r = 0; r < 8; ++r) {
        float* srow = sim + (size_t)(mrow + r) * KCOLS + colw;
        srow[0]  = c0[r];
        srow[16] = c1[r];
        srow[32] = c2[r];
        srow[48] = c3[r];
    }
}

// ---------------------------------------------------------------------------
// Per-row top-100 (iterative argmax, tie -> smallest index, matching
// jax.lax.top_k ordering) + gather of qT rows into the output.
// One 256-thread block per sim row; the 32KB row lives in LDS.
// ---------------------------------------------------------------------------
__global__ __launch_bounds__(256) void knn_topk_gather(const float* __restrict__ sim,
                                                       const float* __restrict__ qT,
                                                       float* __restrict__ out_feat,
                                                       int* __restrict__ out_inds,
                                                       int row_base) {
    __shared__ float vals[KCOLS];             // 32KB
    __shared__ float redv[256];
    __shared__ int   redi[256];
    __shared__ int   topIdx[TOPK + 12];

    const int tid = threadIdx.x;
    const int row_local = blockIdx.x;
    const size_t grow = (size_t)row_base + row_local;

    // Stage the sim row into LDS (float4 vectorized).
    {
        const float4* s4 = (const float4*)(sim + (size_t)row_local * KCOLS);
        float4* v4 = (float4*)vals;
        const int n4 = KCOLS / 4;             // 2048
        for (int i = tid; i < n4; i += 256) v4[i] = s4[i];
    }
    __syncthreads();

    for (int t = 0; t < TOPK; ++t) {
        // Per-thread scan over strided partition (ascending indices, so strict
        // '>' keeps the smallest index among equal values within a thread).
        float best = -__builtin_inff();
        int   bi   = KCOLS;
        for (int j = tid; j < KCOLS; j += 256) {
            float v = vals[j];
            if (v > best) { best = v; bi = j; }
        }
        redv[tid] = best;
        redi[tid] = bi;
        __syncthreads();
        // Tree reduce with jax tie-break: higher value, then lower index.
#pragma unroll
        for (int s = 128; s > 0; s >>= 1) {
            if (tid < s) {
                float ov = redv[tid + s]; int oi = redi[tid + s];
                float mv = redv[tid];     int mi = redi[tid];
                if (ov > mv || (ov == mv && oi < mi)) { redv[tid] = ov; redi[tid] = oi; }
            }
            __syncthreads();
        }
        if (tid == 0) {
            int w = redi[0];
            topIdx[t] = w;
            vals[w] = -__builtin_inff();      // exclude winner
        }
        __syncthreads();
    }

    // Emit indices (int32 block appended after the float feature block).
    if (tid < TOPK) out_inds[grow * TOPK + tid] = topIdx[tid];

    // Gather qT rows: each of 8 waves owns every 8th selected index; each lane
    // copies 4 float4s (512 floats per row, fully coalesced 2KB bursts).
    const int wave = tid >> 5;
    const int lane = tid & 31;
    for (int j = wave; j < TOPK; j += WAVES) {
        const float4* src = (const float4*)(qT + (size_t)topIdx[j] * DIM);
        float4* dst = (float4*)(out_feat + (grow * TOPK + j) * DIM);
#pragma unroll
        for (int i = 0; i < DIM / 4 / 32; ++i)   // 4 iterations
            dst[lane + i * 32] = src[lane + i * 32];
    }
}

// ---------------------------------------------------------------------------
extern "C" void kernel_launch(void* const* d_in, const int* in_sizes, int n_in,
                              void* d_out, int out_size, void* d_ws, size_t ws_size,
                              hipStream_t stream) {
    const float* x = (const float*)d_in[0];       // [16,512,512] -> [8192,512]
    const float* q = (const float*)d_in[1];       // [512,8192]

    float* out_feat = (float*)d_out;                                 // [8192,100,512]
    int*   out_inds = (int*)(out_feat + (size_t)NROWS * TOPK * DIM); // [8192,100]

    // Workspace layout: qT (16MB) then sim chunk.
    float* qT  = (float*)d_ws;
    float* sim = qT + (size_t)KCOLS * DIM;
    const size_t qT_bytes = (size_t)KCOLS * DIM * sizeof(float);

    // Adaptive chunk: as many rows of sim as ws allows, up to 512 (16MB, L2-resident).
    int chunk = 512;
    if (ws_size > qT_bytes) {
        size_t avail_rows = (ws_size - qT_bytes) / ((size_t)KCOLS * sizeof(float));
        if (avail_rows < (size_t)chunk) chunk = (int)(avail_rows & ~(size_t)15);
    }
    if (chunk < 16) chunk = 16;

    // 1) Transpose queue once per call.
    knn_transpose<<<dim3(KCOLS / 32, DIM / 32), dim3(32, 8), 0, stream>>>(q, qT);

    // 2) Chunked GEMM (WMMA f32) + top-k/gather.
    for (int base = 0; base < NROWS; base += chunk) {
        int rows = NROWS - base;
        if (rows > chunk) rows = chunk;
        dim3 ggrid(rows / 16, KCOLS / (16 * WAVES * CTILES));
        knn_gemm<<<ggrid, 256, 0, stream>>>(x, q, sim, base);
        knn_topk_gather<<<rows, 256, 0, stream>>>(sim, qT, out_feat, out_inds, base);
    }
}